// RoIBBox_69097433858702
// MI455X (gfx1250) — compile-verified
//
#include <hip/hip_runtime.h>
#include <hip/hip_bf16.h>

#define TOTAL_A   57600
#define BATCH_N   64
#define PRE_N     6000
#define POST_N    300
#define NEG_VAL   (-1000000000.0f)
#define IOU_THR_F 0.7f

#define TPB_SEL   1024
#define TPB_NMS   512

typedef unsigned int v4u __attribute__((ext_vector_type(4)));
typedef int          v4i __attribute__((ext_vector_type(4)));
typedef int          v8i __attribute__((ext_vector_type(8)));

// Order-preserving float -> uint transform (monotone: a<b  =>  key(a)<key(b))
__device__ __forceinline__ unsigned score_key(float s) {
    unsigned u = __float_as_uint(s);
    return (u & 0x80000000u) ? ~u : (u | 0x80000000u);
}

// ---------------------------------------------------------------------------
// Tensor Data Mover: 1D contiguous DMA global -> LDS.
// D# per CDNA5 ISA ch.8: group0 = {count=1, lds_addr, global_addr[56:0], type=2},
// group1 = {data_size=3 (8B), tensor_dim0=n, tensor_dim1=1, tile_dim0=n,
//           tile_dim1=1, tensor_dim0_stride=n}.  nelem8 must be <= 65535.
// ---------------------------------------------------------------------------
#if __has_builtin(__builtin_amdgcn_tensor_load_to_lds)
#define HAVE_TDM 1
__device__ __forceinline__ void tdm_load_1d(unsigned lds_addr,
                                            unsigned long long gaddr,
                                            unsigned nelem8)
{
    v4u g0;
    g0.x = 1u;                                                  // count=1 (user D#)
    g0.y = lds_addr;                                            // lds_addr[31:0]
    g0.z = (unsigned)(gaddr & 0xffffffffull);                   // global_addr[31:0]
    g0.w = (unsigned)((gaddr >> 32) & 0x01ffffffull) |          // global_addr[56:32]
           (2u << 30);                                          // type=2 ("image")
    v8i g1;
    g1[0] = (int)(3u << 16);                                    // data_size=3 (8B), mask=0
    g1[1] = (int)((nelem8 & 0xffffu) << 16);                    // tensor_dim0[15:0]
    g1[2] = (int)(((nelem8 >> 16) & 0xffffu) | (1u << 16));     // dim0[31:16], tensor_dim1=1
    g1[3] = (int)((nelem8 & 0xffffu) << 16);                    // tile_dim0 = nelem8
    g1[4] = 1;                                                  // tile_dim1=1, tile_dim2=0
    g1[5] = (int)nelem8;                                        // tensor_dim0_stride[31:0]
    g1[6] = 0;                                                  // stride hi, dim1_stride lo
    g1[7] = 0;
    v4i z4 = {0, 0, 0, 0};
#if __has_include(<hip/amd_detail/amd_gfx1250_TDM.h>)
    v8i z8 = {0, 0, 0, 0, 0, 0, 0, 0};                          // 6-arg toolchain
    __builtin_amdgcn_tensor_load_to_lds(g0, g1, z4, z4, z8, 0);
#else
    __builtin_amdgcn_tensor_load_to_lds(g0, g1, z4, z4, 0);     // 5-arg ROCm 7.2
#endif
}
#endif

// ---------------------------------------------------------------------------
// Kernel 1: exact per-image top-PRE_N selection (3-level radix select on the
// order-preserving key: 12 + 12 + 8 bits) + deterministic ordered compaction
// + bbox delta decode on the survivors.  The full per-image score vector
// (230.4 KB) is parked in LDS via async global->LDS copies, so all five
// passes run at LDS speed.
// ---------------------------------------------------------------------------
__global__ __launch_bounds__(TPB_SEL, 1)
void topk_decode_kernel(const float* __restrict__ deltas,
                        const float* __restrict__ scores,
                        const float* __restrict__ anchors,
                        float4* __restrict__ topBoxes,
                        float* __restrict__ topScores)
{
    __shared__ float    scLDS[TOTAL_A];      // 230,400 B
    __shared__ int      hist[4096];          //  16,384 B
    __shared__ int      scanA[TPB_SEL];      //   4,096 B
    __shared__ int      scanB[TPB_SEL];      //   4,096 B
    __shared__ int      s_chi;               // count of keys strictly above threshold
    __shared__ unsigned s_thresh;            // accumulated radix threshold

    const int img = blockIdx.x;
    const int tid = threadIdx.x;
    const float* sc = scores + (size_t)img * TOTAL_A;

    // --- async fill of the score vector into LDS (ASYNCcnt path) ---
    {
        const float4* g4 = (const float4*)sc;
        for (int j = tid; j < TOTAL_A / 4; j += TPB_SEL) {
            unsigned lds = (unsigned)(unsigned long long)(void*)&scLDS[4 * j];
            unsigned long long ga = (unsigned long long)(const void*)(g4 + j);
            asm volatile("global_load_async_to_lds_b128 %0, %1, off"
                         :: "v"(lds), "v"(ga) : "memory");
        }
    }
    if (tid == 0) { s_chi = 0; s_thresh = 0u; }
    asm volatile("s_wait_asynccnt 0" ::: "memory");
    __syncthreads();

    // --- 3-level radix select: find threshold key T and chi = #(key > T) ---
    for (int lvl = 0; lvl < 3; ++lvl) {
        const int sh   = (lvl == 0) ? 20 : (lvl == 1) ? 8 : 0;
        const int nb   = (lvl == 2) ? 256 : 4096;
        const int hish = (lvl == 1) ? 20 : 8;   // bits above this level (lvl>0)

        for (int i = tid; i < nb; i += TPB_SEL) hist[i] = 0;
        __syncthreads();

        const unsigned th = s_thresh;
        for (int idx = tid; idx < TOTAL_A; idx += TPB_SEL) {
            unsigned k = score_key(scLDS[idx]);
            bool in = true;
            if (lvl > 0) in = ((k >> hish) == (th >> hish));
            if (in) atomicAdd(&hist[(k >> sh) & (nb - 1)], 1);
        }
        __syncthreads();

        if (tid == 0) {
            const int need = PRE_N - s_chi;
            int cum = 0, b;
            for (b = nb - 1; b >= 0; --b) { cum += hist[b]; if (cum >= need) break; }
            if (b < 0) b = 0;
            s_chi   += cum - hist[b];
            s_thresh = th | ((unsigned)b << sh);
        }
        __syncthreads();
    }

    const unsigned thresh = s_thresh;
    const int      chi    = s_chi;

    // --- per-thread counts of A (k > T) and B (k == T), then block scan ---
    int cA = 0, cB = 0;
    for (int idx = tid; idx < TOTAL_A; idx += TPB_SEL) {
        unsigned k = score_key(scLDS[idx]);
        cA += (k > thresh);
        cB += (k == thresh);
    }
    scanA[tid] = cA;
    scanB[tid] = cB;
    for (int off = 1; off < TPB_SEL; off <<= 1) {
        __syncthreads();
        int va = 0, vb = 0;
        if (tid >= off) { va = scanA[tid - off]; vb = scanB[tid - off]; }
        __syncthreads();
        scanA[tid] += va;
        scanB[tid] += vb;
    }
    __syncthreads();
    int posA = scanA[tid] - cA;            // exclusive prefix
    int posB = chi + (scanB[tid] - cB);    // B region starts after all A

    // --- write survivors: decode deltas -> boxes, gather scores ---
    const float4* dl = (const float4*)deltas + (size_t)img * TOTAL_A;
    const float4* an = (const float4*)anchors;
    float4* ob = topBoxes  + (size_t)img * PRE_N;
    float*  os = topScores + (size_t)img * PRE_N;

    for (int idx = tid; idx < TOTAL_A; idx += TPB_SEL) {
        float    s = scLDS[idx];
        unsigned k = score_key(s);
        int pos = -1;
        if (k > thresh)       { pos = posA++; }
        else if (k == thresh) { int p = posB++; if (p < PRE_N) pos = p; }
        if (pos >= 0) {
            float4 d = dl[idx];
            float4 a = an[idx];
            float ah  = a.z - a.x;
            float aw  = a.w - a.y;
            float acy = a.x + 0.5f * ah;
            float acx = a.y + 0.5f * aw;
            float bh  = expf(d.z * 0.2f) * ah;        // variance 0.2 on h/w
            float bw  = expf(d.w * 0.2f) * aw;
            float bcy = d.x * 0.1f * ah + acy;        // variance 0.1 on cy/cx
            float bcx = d.y * 0.1f * aw + acx;
            float y1 = bcy - 0.5f * bh;
            float x1 = bcx - 0.5f * bw;
            float4 o;
            o.x = y1; o.y = x1; o.z = y1 + bh; o.w = x1 + bw;
            ob[pos] = o;
            os[pos] = s;
        }
    }
}

// ---------------------------------------------------------------------------
// Kernel 2: greedy NMS, one workgroup per image. Entire candidate set
// (6000 boxes + scores = 120 KB) lives in LDS.  LDS fill: Tensor Data Mover
// (one DMA per tile, TENSORcnt) when the builtin exists, else per-lane async
// global->LDS b128 copies (ASYNCcnt).
// Argmax = wave32 shfl reduce on a packed (score,~idx) u64 key + one
// ds_max u64 LDS atomic per wave; 2 barriers per iteration.
// ---------------------------------------------------------------------------
__global__ __launch_bounds__(TPB_NMS, 1)
void nms_kernel(const float4* __restrict__ topBoxes,
                const float* __restrict__ topScores,
                float4* __restrict__ out)
{
    __shared__ float4 boxesLDS[PRE_N];                 // 96 KB
    __shared__ float  scoresLDS[PRE_N];                // 24 KB
    __shared__ unsigned long long gkey[2];

    const int img = blockIdx.x;
    const int tid = threadIdx.x;
    const float4* gb = topBoxes  + (size_t)img * PRE_N;
    const float*  gs = topScores + (size_t)img * PRE_N;

#ifdef HAVE_TDM
    if (tid < 32) {   // wave 0 issues the two DMAs
        tdm_load_1d((unsigned)(unsigned long long)(void*)&boxesLDS[0],
                    (unsigned long long)(const void*)gb,
                    (unsigned)(PRE_N * sizeof(float4) / 8));    // 12000 x 8B
        tdm_load_1d((unsigned)(unsigned long long)(void*)&scoresLDS[0],
                    (unsigned long long)(const void*)gs,
                    (unsigned)(PRE_N * sizeof(float) / 8));     //  3000 x 8B
        __builtin_amdgcn_s_wait_tensorcnt(0);
    }
#else
    for (int j = tid; j < PRE_N; j += TPB_NMS) {
        unsigned lds = (unsigned)(unsigned long long)(void*)&boxesLDS[j];
        unsigned long long ga = (unsigned long long)(const void*)(gb + j);
        asm volatile("global_load_async_to_lds_b128 %0, %1, off"
                     :: "v"(lds), "v"(ga) : "memory");
    }
    const float4* gs4 = (const float4*)gs;
    for (int j = tid; j < PRE_N / 4; j += TPB_NMS) {
        unsigned lds = (unsigned)(unsigned long long)(void*)&scoresLDS[4 * j];
        unsigned long long ga = (unsigned long long)(const void*)(gs4 + j);
        asm volatile("global_load_async_to_lds_b128 %0, %1, off"
                     :: "v"(lds), "v"(ga) : "memory");
    }
    asm volatile("s_wait_asynccnt 0" ::: "memory");
#endif
    if (tid < 2) gkey[tid] = 0ull;
    __syncthreads();

    for (int i = 0; i < POST_N; ++i) {
        const int k = i & 1;

        // local argmax over this thread's strided partition
        unsigned long long key = 0ull;
        for (int j = tid; j < PRE_N; j += TPB_NMS) {
            unsigned u = score_key(scoresLDS[j]);
            unsigned long long kk =
                ((unsigned long long)u << 32) | (unsigned)(PRE_N - j); // low idx wins ties
            key = (kk > key) ? kk : key;
        }
        // wave32 reduction, then one LDS u64 max-atomic per wave
        #pragma unroll
        for (int off = 16; off > 0; off >>= 1) {
            unsigned long long o = __shfl_down(key, (unsigned)off, 32);
            key = (o > key) ? o : key;
        }
        if ((tid & 31) == 0) atomicMax(&gkey[k], key);
        __syncthreads();

        const unsigned long long bk = gkey[k];
        const int    bj     = PRE_N - (int)(unsigned)(bk & 0xffffffffull);
        const float4 bb     = boxesLDS[bj];
        const float  bscore = scoresLDS[bj];

        if (tid == 0) {
            gkey[k ^ 1] = 0ull;   // reset ping-pong slot for next iteration
            float4 o;
            if (bscore > NEG_VAL * 0.5f) {
                o.x = fminf(fmaxf(bb.x, 0.0f), 1.0f);
                o.y = fminf(fmaxf(bb.y, 0.0f), 1.0f);
                o.z = fminf(fmaxf(bb.z, 0.0f), 1.0f);
                o.w = fminf(fmaxf(bb.w, 0.0f), 1.0f);
            } else {
                o.x = o.y = o.z = o.w = 0.0f;
            }
            out[(size_t)img * POST_N + i] = o;
        }

        // suppression
        const float barea = (bb.z - bb.x) * (bb.w - bb.y);
        for (int j = tid; j < PRE_N; j += TPB_NMS) {
            float4 c = boxesLDS[j];
            float yy1 = fmaxf(c.x, bb.x);
            float xx1 = fmaxf(c.y, bb.y);
            float yy2 = fminf(c.z, bb.z);
            float xx2 = fminf(c.w, bb.w);
            float inter = fmaxf(yy2 - yy1, 0.0f) * fmaxf(xx2 - xx1, 0.0f);
            float carea = (c.z - c.x) * (c.w - c.y);
            float iou   = inter / (carea + barea - inter + 1e-8f);
            if (iou > IOU_THR_F || j == bj) scoresLDS[j] = NEG_VAL;
        }
        __syncthreads();
    }
}

// ---------------------------------------------------------------------------
extern "C" void kernel_launch(void* const* d_in, const int* in_sizes, int n_in,
                              void* d_out, int out_size, void* d_ws, size_t ws_size,
                              hipStream_t stream) {
    (void)in_sizes; (void)n_in; (void)out_size; (void)ws_size;
    const float* deltas  = (const float*)d_in[0];   // [64, 57600, 4]
    const float* labels  = (const float*)d_in[1];   // [64, 57600]
    const float* anchors = (const float*)d_in[2];   // [57600, 4]

    float4* topBoxes  = (float4*)d_ws;                                      // 64*6000*16 B
    float*  topScores = (float*)((char*)d_ws +
                                 (size_t)BATCH_N * PRE_N * sizeof(float4)); // 64*6000*4 B

    topk_decode_kernel<<<BATCH_N, TPB_SEL, 0, stream>>>(
        deltas, labels, anchors, topBoxes, topScores);
    nms_kernel<<<BATCH_N, TPB_NMS, 0, stream>>>(
        topBoxes, topScores, (float4*)d_out);
}